// SemiSupervisedSAGELayer_43499428774649
// MI455X (gfx1250) — compile-verified
//
#include <hip/hip_runtime.h>
#include <hip/hip_bf16.h>

typedef __attribute__((ext_vector_type(2))) float v2f;
typedef __attribute__((ext_vector_type(8))) float v8f;

// ---------------------------------------------------------------------------
// Kernel 1: zero the workspace (neigh_sum [N,64] + degree [N]) every launch
// ---------------------------------------------------------------------------
__global__ void sage_zero_f32(float* __restrict__ p, size_t n) {
    size_t i = (size_t)blockIdx.x * blockDim.x + threadIdx.x;
    size_t stride = (size_t)gridDim.x * blockDim.x;
    for (; i < n; i += stride) p[i] = 0.0f;
}

// ---------------------------------------------------------------------------
// Kernel 2: edge aggregation. One edge per 16-lane group; each lane owns a
// contiguous float4 of the 64-wide feature row. Scatter-add with f32 atomics
// into the L2-resident workspace (51 MB total << 192 MB L2).
// ---------------------------------------------------------------------------
__global__ void sage_aggregate(const long long* __restrict__ src,
                               const long long* __restrict__ dst,
                               const float*     __restrict__ feat,
                               float*           __restrict__ nsum,
                               float*           __restrict__ deg,
                               int E) {
    size_t gid = (size_t)blockIdx.x * blockDim.x + threadIdx.x;
    int e = (int)(gid >> 4);
    int l = (int)(gid & 15);
    if (e >= E) return;
    int s = (int)src[e];
    int d = (int)dst[e];
    const float4 v = *(const float4*)(feat + (size_t)s * 64 + l * 4);
    float* p = nsum + (size_t)d * 64 + l * 4;
    atomicAdd(p + 0, v.x);
    atomicAdd(p + 1, v.y);
    atomicAdd(p + 2, v.z);
    atomicAdd(p + 3, v.w);
    if (l == 0) atomicAdd(deg + d, 1.0f);
}

// ---------------------------------------------------------------------------
// Kernel 3: out[N,64] = concat(feat, nsum/deg) @ W[128,64] + b  via
// V_WMMA_F32_16X16X4_F32 (full fp32, matches reference precision).
// One wave per 16x16 output tile; K=128 consumed in 32 WMMA steps of 4.
//
// ISA layouts (CDNA5 ISA):
//   A 16x4 f32 : lanes 0-15 -> M=lane, {v0,v1}={K0,K1}; lanes 16-31 -> {K2,K3}
//   B 4x16 f32 : lanes hold N=lane&15; lane-half selects K pair like A
//   C/D 16x16  : vgpr i: lanes 0-15 -> M=i, N=lane; lanes 16-31 -> M=8+i
// ---------------------------------------------------------------------------
__global__ void sage_gemm_wmma(const float* __restrict__ feat,
                               const float* __restrict__ nsum,
                               const float* __restrict__ deg,
                               const float* __restrict__ W,     // [128,64]
                               const float* __restrict__ bias,  // [64]
                               float*       __restrict__ out,   // [N,64]
                               int N) {
    const int lane = threadIdx.x & 31;
    const int wave = blockIdx.x * (blockDim.x >> 5) + (threadIdx.x >> 5);
    const int tm = wave >> 2;        // row tile (64/16 = 4 col tiles)
    const int tn = wave & 3;         // col tile
    const int row0 = tm * 16;
    if (row0 >= N) return;
    const int col0 = tn * 16;

    const int lr = lane & 15;        // M index (A) / N index (B, C/D)
    const int kh = lane >> 4;        // 0 -> K pair {0,1}; 1 -> K pair {2,3}

    const int arow = row0 + lr;      // the A-matrix row this lane carries
    const float invd = 1.0f / fmaxf(deg[arow], 1.0f);

    const float* aSelf = feat + (size_t)arow * 64;
    const float* aNei  = nsum + (size_t)arow * 64;

    v8f acc = {};

    // Self half: feat @ W[0:64, :]
    #pragma unroll
    for (int k4 = 0; k4 < 64; k4 += 4) {
        const int ka = k4 + kh * 2;
        v2f a;  a.x = aSelf[ka];  a.y = aSelf[ka + 1];
        v2f bm; bm.x = W[(size_t)(ka)     * 64 + col0 + lr];
                bm.y = W[(size_t)(ka + 1) * 64 + col0 + lr];
        acc = __builtin_amdgcn_wmma_f32_16x16x4_f32(
            false, a, false, bm, (short)0, acc, false, false);
    }
    // Neighbor half: (nsum * invd) @ W[64:128, :]  (invd folded into A rows)
    #pragma unroll
    for (int k4 = 0; k4 < 64; k4 += 4) {
        const int ka = k4 + kh * 2;
        v2f a;  a.x = aNei[ka] * invd;  a.y = aNei[ka + 1] * invd;
        v2f bm; bm.x = W[(size_t)(64 + ka)     * 64 + col0 + lr];
                bm.y = W[(size_t)(64 + ka + 1) * 64 + col0 + lr];
        acc = __builtin_amdgcn_wmma_f32_16x16x4_f32(
            false, a, false, bm, (short)0, acc, false, false);
    }

    // Write D + bias. Lane-half selects M rows 0-7 vs 8-15.
    const int cn = col0 + lr;
    const float bv = bias[cn];
    const int mb = row0 + (kh ? 8 : 0);
    #pragma unroll
    for (int i = 0; i < 8; i++)
        out[(size_t)(mb + i) * 64 + cn] = acc[i] + bv;
}

// ---------------------------------------------------------------------------
// Host launcher
// ---------------------------------------------------------------------------
extern "C" void kernel_launch(void* const* d_in, const int* in_sizes, int n_in,
                              void* d_out, int out_size, void* d_ws, size_t ws_size,
                              hipStream_t stream) {
    const float*     feat = (const float*)d_in[0];        // [N,64] f32
    const long long* ei   = (const long long*)d_in[1];    // [2,E] i64
    const float*     W    = (const float*)d_in[2];        // [128,64] f32
    const float*     bias = (const float*)d_in[3];        // [64] f32
    float*           out  = (float*)d_out;                // [N,64] f32

    const int D = 64;
    const int N = in_sizes[0] / D;
    const int E = in_sizes[1] / 2;

    // Workspace layout: nsum [N,64] f32, then degree [N] f32 (~25.9 MB)
    float* nsum = (float*)d_ws;
    float* deg  = nsum + (size_t)N * D;

    const size_t zn = (size_t)N * D + (size_t)N;
    sage_zero_f32<<<2048, 256, 0, stream>>>(nsum, zn);

    const size_t aggThreads = (size_t)E * 16;
    const int aggBlocks = (int)((aggThreads + 255) / 256);
    sage_aggregate<<<aggBlocks, 256, 0, stream>>>(ei, ei + E, feat, nsum, deg, E);

    const int tilesM = (N + 15) / 16;
    const int waves  = tilesM * 4;          // 4 col tiles of 16 over D_OUT=64
    const int blocks = (waves + 7) / 8;     // 8 waves per 256-thread block
    sage_gemm_wmma<<<blocks, 256, 0, stream>>>(feat, nsum, deg, W, bias, out, N);
}